// HybridLayer_65481071409275
// MI455X (gfx1250) — compile-verified
//
#include <hip/hip_runtime.h>
#include <math.h>

// out[b] = <Z_0> of RY-layer + CZ-chain + RZ-layer on |0000>  ==  cos(theta[b,0]).
// (CZ signs and RZ phases cancel in |psi|^2; all non-measured qubits contribute
//  cos^2+sin^2 = 1.)  Memory-bound streaming kernel: 32 MB read + 8 MB write
// -> ~1.7 us at 23.3 TB/s; compute (one cos per element) is negligible.
//
// CDNA5 path exercised: GLOBAL_LOAD_ASYNC_TO_LDS_B32 (ASYNCcnt) staging of the
// strided column-0 gather, s_wait_asynccnt, LDS readback, nontemporal store.

#define BLOCK_THREADS   256
#define ROWS_PER_THREAD 4
#define TILE_ROWS       (BLOCK_THREADS * ROWS_PER_THREAD)  // 1024 rows / block

typedef __attribute__((address_space(1))) int as1_int;  // global
typedef __attribute__((address_space(3))) int as3_int;  // LDS

__global__ __launch_bounds__(BLOCK_THREADS) void
hybrid_z0_cos_kernel(const float* __restrict__ in, float* __restrict__ out, int nrows) {
#if defined(__HIP_DEVICE_COMPILE__)
    __shared__ float xbuf[TILE_ROWS];

    const int tid = (int)threadIdx.x;
    const long long base = (long long)blockIdx.x * TILE_ROWS;

    // ---- Stage column 0 (x component of each (B,4) row) into LDS via async copies.
    // Each lane writes only its own LDS slots, so per-wave ASYNCcnt ordering is
    // sufficient and no workgroup barrier is required.
#pragma unroll
    for (int i = 0; i < ROWS_PER_THREAD; ++i) {
        const long long row = base + (long long)(i * BLOCK_THREADS + tid);
        if (row < nrows) {
            const float* gsrc = in + (row << 2);  // element (row, 0); 16 B row stride
#if __has_builtin(__builtin_amdgcn_global_load_async_to_lds_b32)
            __builtin_amdgcn_global_load_async_to_lds_b32(
                (as1_int*)(uintptr_t)gsrc,
                (as3_int*)(unsigned)(uintptr_t)&xbuf[i * BLOCK_THREADS + tid],
                /*imm offset*/ 0, /*cpol*/ 0);
#else
            const unsigned lds_off = (unsigned)(uintptr_t)&xbuf[i * BLOCK_THREADS + tid];
            const unsigned voff    = (unsigned)((uintptr_t)gsrc - (uintptr_t)in);
            // GVS mode: mem_addr = SGPR64 + VGPR32 offset; VDST VGPR holds LDS byte addr.
            asm volatile("global_load_async_to_lds_b32 %0, %1, %2"
                         :: "v"(lds_off), "v"(voff), "s"(in)
                         : "memory");
#endif
        }
    }

    // ---- Wait for this wave's async copies to land in LDS.
#if __has_builtin(__builtin_amdgcn_s_wait_asynccnt)
    __builtin_amdgcn_s_wait_asynccnt(0);
#else
    asm volatile("s_wait_asynccnt 0x0" ::: "memory");
#endif

    // ---- Compute cos(theta0) and stream out (non-temporal: no reuse).
#pragma unroll
    for (int i = 0; i < ROWS_PER_THREAD; ++i) {
        const long long row = base + (long long)(i * BLOCK_THREADS + tid);
        if (row < nrows) {
            const float th = xbuf[i * BLOCK_THREADS + tid];
            __builtin_nontemporal_store(cosf(th), &out[row]);
        }
    }
#else
    (void)in; (void)out; (void)nrows;
#endif
}

extern "C" void kernel_launch(void* const* d_in, const int* in_sizes, int n_in,
                              void* d_out, int out_size, void* d_ws, size_t ws_size,
                              hipStream_t stream) {
    (void)in_sizes; (void)n_in; (void)d_ws; (void)ws_size;
    const float* in = (const float*)d_in[0];   // (B, 4) float32, row-major
    float* out = (float*)d_out;                // (B,) float32
    const int nrows = out_size;                // B = 2^21
    const int grid = (nrows + TILE_ROWS - 1) / TILE_ROWS;
    hybrid_z0_cos_kernel<<<grid, BLOCK_THREADS, 0, stream>>>(in, out, nrows);
}